// SAGE_8014408974454
// MI455X (gfx1250) — compile-verified
//
#include <hip/hip_runtime.h>
#include <hip/hip_bf16.h>

typedef __attribute__((ext_vector_type(2))) float v2f;
typedef __attribute__((ext_vector_type(8))) float v8f;

#define DFEAT 128
#define LN_EPS 1e-5f

// ---------------------------------------------------------------------------
// Zero a float buffer (grid-stride).
__global__ void sage_zero_kernel(float* __restrict__ p, size_t n) {
  size_t i = (size_t)blockIdx.x * blockDim.x + threadIdx.x;
  size_t stride = (size_t)gridDim.x * blockDim.x;
  for (; i < n; i += stride) p[i] = 0.0f;
}

// ---------------------------------------------------------------------------
// Degree count: deg[dst] += 1  (stored as float, one thread per edge)
__global__ void sage_degree_kernel(const int* __restrict__ edst,
                                   float* __restrict__ deg, int n_edges) {
  int e = blockIdx.x * blockDim.x + threadIdx.x;
  if (e < n_edges) atomicAdd(&deg[edst[e]], 1.0f);
}

// deg -> 1/(deg+1) in place
__global__ void sage_invdeg_kernel(float* __restrict__ deg, int n_nodes) {
  int i = blockIdx.x * blockDim.x + threadIdx.x;
  if (i < n_nodes) deg[i] = 1.0f / (deg[i] + 1.0f);
}

// ---------------------------------------------------------------------------
// Scatter-add: accum[dst][:] += feat[src][:].  One wave (32 lanes) per edge,
// 4 floats per lane -> coalesced 512B row gather + global_atomic_add_f32.
__global__ void sage_scatter_kernel(const float* __restrict__ feat,
                                    const int* __restrict__ esrc,
                                    const int* __restrict__ edst,
                                    float* __restrict__ accum, int n_edges) {
  long long t = (long long)blockIdx.x * blockDim.x + threadIdx.x;
  int e = (int)(t >> 5);
  int lane = (int)(t & 31);
  if (e >= n_edges) return;
  int s = esrc[e];
  int d = edst[e];
  const float4 v =
      *reinterpret_cast<const float4*>(feat + (size_t)s * DFEAT + lane * 4);
  float* o = accum + (size_t)d * DFEAT + lane * 4;
  atomicAdd(o + 0, v.x);
  atomicAdd(o + 1, v.y);
  atomicAdd(o + 2, v.z);
  atomicAdd(o + 3, v.w);
}

// ---------------------------------------------------------------------------
// Fused SAGE layer: agg = (neigh + h_self)*inv_deg1 ; lin = agg @ W + b ;
// optionally LN+ReLU -> out_act.  One block (8 wave32) per 16-node tile.
// Each wave computes a 16x16 output tile via 32 x v_wmma_f32_16x16x4_f32.
__global__ __launch_bounds__(256) void sage_layer_kernel(
    const float* __restrict__ hin,     // [N,128] self features
    const float* __restrict__ neigh,   // [N,128] summed neighbor features
    const float* __restrict__ invd,    // [N]     1/(deg+1)
    const float* __restrict__ W,       // [128,128] row-major (k,n)
    const float* __restrict__ bias,    // [128]
    const float* __restrict__ gamma,   // [128] (unused if !do_ln)
    const float* __restrict__ beta,    // [128] (unused if !do_ln)
    float* __restrict__ out_lin,       // [N,128] pre-norm linear output
    float* __restrict__ out_act,       // [N,128] relu(LN(lin)) (if do_ln)
    int n_nodes, int do_ln) {
  __shared__ float Asm[16][132];  // padded: conflict-free A-fragment reads

  const int base = blockIdx.x * 16;
  const int tid = threadIdx.x;

  // Stage agg tile (16 nodes x 128 feats) into LDS.
  for (int idx = tid; idx < 16 * DFEAT; idx += 256) {
    int r = idx >> 7;
    int c = idx & (DFEAT - 1);
    int node = base + r;
    if (node >= n_nodes) node = n_nodes - 1;
    float s = neigh[(size_t)node * DFEAT + c] + hin[(size_t)node * DFEAT + c];
    Asm[r][c] = s * invd[node];
  }
  __syncthreads();

  const int wave = tid >> 5;           // 8 waves, wave32
  const int lane = tid & 31;
  const int m = lane & 15;             // A: row M / B,D: col N (mod 16)
  const int kb = (lane < 16) ? 0 : 2;  // K-pair select per lane half
  const int n = wave * 16 + m;         // output column owned by this lane

  v8f acc = {0.f, 0.f, 0.f, 0.f, 0.f, 0.f, 0.f, 0.f};
#pragma unroll 8
  for (int k0 = 0; k0 < DFEAT; k0 += 4) {
    v2f a, b;
    // 32-bit A 16x4 layout: lane M=lane&15; VGPR0/1 = K {0,1} (lanes 0-15)
    // or K {2,3} (lanes 16-31).
    a.x = Asm[m][k0 + kb];
    a.y = Asm[m][k0 + kb + 1];
    // B 4x16 (K x N): lane N=lane&15, same K-pair split.
    b.x = W[(size_t)(k0 + kb) * DFEAT + n];
    b.y = W[(size_t)(k0 + kb + 1) * DFEAT + n];
    acc = __builtin_amdgcn_wmma_f32_16x16x4_f32(false, a, false, b, (short)0,
                                                acc, false, false);
  }

  if (do_ln) __syncthreads();  // all waves done reading Asm before reuse

  // D layout: VGPR j -> M = j + (lane<16 ? 0 : 8), N = lane&15.
  const float bn = bias[n];
  const int rbase = (lane < 16) ? 0 : 8;
#pragma unroll
  for (int j = 0; j < 8; ++j) {
    int r = rbase + j;
    int node = base + r;
    float v = acc[j] + bn;
    if (node < n_nodes) out_lin[(size_t)node * DFEAT + n] = v;
    if (do_ln) Asm[r][n] = v;  // stage for layernorm
  }

  if (do_ln) {
    __syncthreads();
    // LayerNorm + ReLU: each wave handles 2 rows; 4 cols per lane.
#pragma unroll
    for (int rr = 0; rr < 2; ++rr) {
      int r = wave * 2 + rr;
      int node = base + r;
      int c = lane << 2;
      const float4 x = *reinterpret_cast<const float4*>(&Asm[r][c]);
      float s = x.x + x.y + x.z + x.w;
#pragma unroll
      for (int off = 16; off > 0; off >>= 1) s += __shfl_xor(s, off, 32);
      float mu = s * (1.0f / DFEAT);
      float d0 = x.x - mu, d1 = x.y - mu, d2 = x.z - mu, d3 = x.w - mu;
      float q = d0 * d0 + d1 * d1 + d2 * d2 + d3 * d3;
#pragma unroll
      for (int off = 16; off > 0; off >>= 1) q += __shfl_xor(q, off, 32);
      float inv = rsqrtf(q * (1.0f / DFEAT) + LN_EPS);
      if (node < n_nodes) {
        float4 o;
        o.x = fmaxf(0.f, gamma[c + 0] * d0 * inv + beta[c + 0]);
        o.y = fmaxf(0.f, gamma[c + 1] * d1 * inv + beta[c + 1]);
        o.z = fmaxf(0.f, gamma[c + 2] * d2 * inv + beta[c + 2]);
        o.w = fmaxf(0.f, gamma[c + 3] * d3 * inv + beta[c + 3]);
        *reinterpret_cast<float4*>(&out_act[(size_t)node * DFEAT + c]) = o;
      }
    }
  }
}

// ---------------------------------------------------------------------------
extern "C" void kernel_launch(void* const* d_in, const int* in_sizes, int n_in,
                              void* d_out, int out_size, void* d_ws,
                              size_t ws_size, hipStream_t stream) {
  const float* feats = (const float*)d_in[0];
  const int* esrc = (const int*)d_in[1];
  const int* edst = (const int*)d_in[2];
  const float* W0 = (const float*)d_in[3];
  const float* b0 = (const float*)d_in[4];
  const float* g0 = (const float*)d_in[5];
  const float* be0 = (const float*)d_in[6];
  const float* W1 = (const float*)d_in[7];
  const float* b1 = (const float*)d_in[8];

  const int N = in_sizes[0] / DFEAT;
  const int E = in_sizes[1];

  float* out = (float*)d_out;                  // [N,128] h_hidden
  float* out2 = out + (size_t)N * DFEAT;       // [N,128] final
  float* ws = (float*)d_ws;
  const size_t degPad = ((size_t)N + 255) & ~(size_t)255;
  float* invdeg = ws;                          // N floats
  float* neighb = ws + degPad;                 // N*128 floats (reused)
  float* hrelu = neighb + (size_t)N * DFEAT;   // N*128 floats

  const int ZB = 4096;
  // 1) zero deg + neighbor accumulator
  sage_zero_kernel<<<ZB, 256, 0, stream>>>(invdeg,
                                           degPad + (size_t)N * DFEAT);
  // 2) degrees -> inv(deg+1)
  sage_degree_kernel<<<(E + 255) / 256, 256, 0, stream>>>(edst, invdeg, E);
  sage_invdeg_kernel<<<(N + 255) / 256, 256, 0, stream>>>(invdeg, N);
  // 3) scatter feats into neighbor sums
  long long sthreads = (long long)E * 32;
  int sblocks = (int)((sthreads + 255) / 256);
  sage_scatter_kernel<<<sblocks, 256, 0, stream>>>(feats, esrc, edst, neighb,
                                                   E);
  // 4) layer 0: agg -> W0 gemm -> h_hidden (d_out) + LN/ReLU -> hrelu
  sage_layer_kernel<<<(N + 15) / 16, 256, 0, stream>>>(
      feats, neighb, invdeg, W0, b0, g0, be0, out, hrelu, N, 1);
  // 5) re-zero accumulator, scatter hrelu
  sage_zero_kernel<<<ZB, 256, 0, stream>>>(neighb, (size_t)N * DFEAT);
  sage_scatter_kernel<<<sblocks, 256, 0, stream>>>(hrelu, esrc, edst, neighb,
                                                   E);
  // 6) layer 1: agg -> W1 gemm -> final output (no LN)
  sage_layer_kernel<<<(N + 15) / 16, 256, 0, stream>>>(
      hrelu, neighb, invdeg, W1, b1, nullptr, nullptr, out2, nullptr, N, 0);
}